// RGCN_41266045780250
// MI455X (gfx1250) — compile-verified
//
#include <hip/hip_runtime.h>

// Problem constants (match reference)
#define NN 100000   // nodes
#define RR 4        // relations
#define DD 64       // feature dim (in == out)

typedef __attribute__((ext_vector_type(2))) float v2f;
typedef __attribute__((ext_vector_type(8))) float v8f;

// ---------------------------------------------------------------- utilities
__global__ void rgcn_fill_zero(float* __restrict__ p, int n) {
  int i = blockIdx.x * blockDim.x + threadIdx.x;
  if (i < n) p[i] = 0.0f;
}

__global__ void rgcn_count_deg(const int* __restrict__ et,
                               const int* __restrict__ dst,
                               float* __restrict__ deg, int E) {
  int e = blockIdx.x * blockDim.x + threadIdx.x;
  if (e < E) {
    int seg = et[e] * NN + dst[e];
    unsafeAtomicAdd(&deg[seg], 1.0f);
  }
}

__global__ void rgcn_to_norm(float* __restrict__ deg, int n) {
  int i = blockIdx.x * blockDim.x + threadIdx.x;
  if (i < n) {
    float d = deg[i];
    deg[i] = (d > 0.5f) ? (1.0f / d) : 0.0f;
  }
}

__global__ void rgcn_relu(float* __restrict__ p, int n) {
  int i = blockIdx.x * blockDim.x + threadIdx.x;
  if (i < n) p[i] = fmaxf(p[i], 0.0f);
}

// ---------------------------------------------------------------- WMMA GEMM
// One wave owns a 16-row strip of X and iterates over all 20 column tiles:
//   tiles 0..15  -> H[r][m][0..63] = X @ W[r]   (r = tile/4)
//   tiles 16..19 -> base[m][0..63] = X @ root + bias
// A fragments (K=64 -> 16 x v2f = 32 VGPRs) are loaded ONCE and reused for
// all 20 tiles, so X is read exactly once per layer. 16 WMMAs per tile.
__global__ void __launch_bounds__(128)
rgcn_gemm_wmma(const float* __restrict__ X,
               const float* __restrict__ W,     // [R][64][64]
               const float* __restrict__ root,  // [64][64]
               const float* __restrict__ bias,  // [64]
               float* __restrict__ H,           // [R][N][64]
               float* __restrict__ base,        // [N][64]
               int nrows) {
  const int wave = threadIdx.x >> 5;   // 0..3
  const int lane = threadIdx.x & 31;
  const int half = lane >> 4;          // 0: lanes 0-15, 1: lanes 16-31
  const int lr   = lane & 15;

  const int m0 = (blockIdx.x * 4 + wave) * 16;
  if (m0 >= nrows) return;             // wave-uniform: EXEC stays all-ones

  // A[16x4] fragments for the whole K=64: lane holds A[lr][4kk + 2*half + {0,1}]
  v2f a[16];
  {
    const float* arow = X + (size_t)(m0 + lr) * DD + 2 * half;
#pragma unroll
    for (int kk = 0; kk < 16; ++kk) {
      a[kk].x = arow[4 * kk];
      a[kk].y = arow[4 * kk + 1];
    }
  }

  const float bv = bias[lr];           // bias element for this lane's column

#pragma unroll 1
  for (int nt = 0; nt < 20; ++nt) {
    const float* Wp;
    int nb;
    if (nt < 16) { Wp = W + (size_t)(nt >> 2) * (DD * DD); nb = (nt & 3) * 16; }
    else         { Wp = root;                               nb = (nt - 16) * 16; }

    // B[4x16] fragment: lane holds B[4kk + 2*half + {0,1}][nb + lr]
    const float* bcol = Wp + (size_t)(2 * half) * DD + nb + lr;

    v8f acc = {};
#pragma unroll
    for (int kk = 0; kk < 16; ++kk) {
      v2f b;
      b.x = bcol[(size_t)(4 * kk) * DD];
      b.y = bcol[(size_t)(4 * kk + 1) * DD];
      acc = __builtin_amdgcn_wmma_f32_16x16x4_f32(
          /*neg_a=*/false, a[kk], /*neg_b=*/false, b,
          /*c_mod=*/(short)0, acc, /*reuse_a=*/false, /*reuse_b=*/false);
    }

    // C/D layout: acc[i] = D[i + 8*half][lr] of the 16x16 tile
    if (nt < 16) {
      const int r = nt >> 2;
      float* Hp = H + ((size_t)r * nrows + m0 + 8 * half) * DD + nb + lr;
#pragma unroll
      for (int i = 0; i < 8; ++i) Hp[(size_t)i * DD] = acc[i];
    } else {
      // note: column of this lane is nb + lr; bias indexed accordingly
      const float bvc = (nb == 0 && false) ? 0.f : bias[nb + lr];
      float* Bp = base + (size_t)(m0 + 8 * half) * DD + nb + lr;
#pragma unroll
      for (int i = 0; i < 8; ++i) Bp[(size_t)i * DD] = acc[i] + bvc;
    }
  }
  (void)bv;
}

// ---------------------------------------------------------------- aggregate
// 16 lanes per edge, float4 per lane:
//   out[dst][t..t+3] += H[etype][src][t..t+3] * norm[etype*N + dst]
__global__ void rgcn_aggregate(const int* __restrict__ src,
                               const int* __restrict__ dst,
                               const int* __restrict__ et,
                               const float* __restrict__ norm,
                               const float* __restrict__ H,
                               float* __restrict__ out, int E) {
  long long gid = (long long)blockIdx.x * blockDim.x + threadIdx.x;
  int e = (int)(gid >> 4);
  int t = ((int)gid & 15) * 4;
  if (e >= E) return;
  int r = et[e];
  int s = src[e];
  int d = dst[e];
  float nrm = norm[(size_t)r * NN + d];
  const float4 hv = *(const float4*)(H + ((size_t)r * NN + s) * DD + t);
  float* op = out + (size_t)d * DD + t;
  unsafeAtomicAdd(op + 0, hv.x * nrm);
  unsafeAtomicAdd(op + 1, hv.y * nrm);
  unsafeAtomicAdd(op + 2, hv.z * nrm);
  unsafeAtomicAdd(op + 3, hv.w * nrm);
}

// ---------------------------------------------------------------- launcher
extern "C" void kernel_launch(void* const* d_in, const int* in_sizes, int n_in,
                              void* d_out, int out_size, void* d_ws, size_t ws_size,
                              hipStream_t stream) {
  const float* x     = (const float*)d_in[0];
  const float* W1    = (const float*)d_in[1];
  const float* root1 = (const float*)d_in[2];
  const float* b1    = (const float*)d_in[3];
  const float* W2    = (const float*)d_in[4];
  const float* root2 = (const float*)d_in[5];
  const float* b2    = (const float*)d_in[6];
  const int*   ei    = (const int*)d_in[7];   // [2][E]
  const int*   et    = (const int*)d_in[8];   // [E]
  const int E = in_sizes[8];
  const int* src = ei;
  const int* dst = ei + E;

  // workspace layout (floats): H | norm | h1   (~130 MB total)
  float* H    = (float*)d_ws;                       // R*N*64
  float* norm = H + (size_t)RR * NN * DD;           // R*N
  float* h1   = norm + (size_t)RR * NN;             // N*64
  float* outp = (float*)d_out;

  const int segs = RR * NN;
  const int feat = NN * DD;
  const long long eth = (long long)E * 16;          // 16 lanes per edge

  // degree -> 1/deg (shared by both layers; depends only on edges)
  rgcn_fill_zero<<<(segs + 255) / 256, 256, 0, stream>>>(norm, segs);
  rgcn_count_deg<<<(E + 255) / 256, 256, 0, stream>>>(et, dst, norm, E);
  rgcn_to_norm<<<(segs + 255) / 256, 256, 0, stream>>>(norm, segs);

  dim3 ggrid((NN + 63) / 64);  // 4 waves x 16 rows per block; each wave does 20 tiles

  // Layer 1: H = x@W1 (per relation), h1 = x@root1 + b1, then scatter, ReLU
  rgcn_gemm_wmma<<<ggrid, 128, 0, stream>>>(x, W1, root1, b1, H, h1, NN);
  rgcn_aggregate<<<(int)((eth + 255) / 256), 256, 0, stream>>>(src, dst, et, norm, H, h1, E);
  rgcn_relu<<<(feat + 255) / 256, 256, 0, stream>>>(h1, feat);

  // Layer 2: into d_out (gemm writes every element of d_out before atomics)
  rgcn_gemm_wmma<<<ggrid, 128, 0, stream>>>(h1, W2, root2, b2, H, outp, NN);
  rgcn_aggregate<<<(int)((eth + 255) / 256), 256, 0, stream>>>(src, dst, et, norm, H, outp, E);
}